// SP_upSampling_38062000177827
// MI455X (gfx1250) — compile-verified
//
#include <hip/hip_runtime.h>

typedef float v2f __attribute__((ext_vector_type(2)));
typedef float v8f __attribute__((ext_vector_type(8)));

// Problem constants (match reference): B=4, C=64, G=64, P=8
#define LDST 68   // LDS row stride in floats: 68*4 = 272 B = multiple of 16 B

__global__ __launch_bounds__(128) void sp_upsample_wmma(
    const float* __restrict__ fea,    // [4, 4096, 64]
    const float* __restrict__ prob,   // [4, 9, 512, 512]
    float* __restrict__ out)          // [4, 64, 512, 512]
{
    __shared__ float attnS[12 * LDST];   // [k][p], K padded 9 -> 12
    __shared__ float kvmS [12 * LDST];   // [k][c]
    __shared__ float outS [64 * LDST];   // [c][p]

    const int tid  = threadIdx.x;
    const int cell = blockIdx.x;          // b*4096 + gy*64 + gx
    const int b    = cell >> 12;
    const int gy   = (cell >> 6) & 63;
    const int gx   = cell & 63;

    // ---- stage attn: prob[b, k, gy*8+py, gx*8+px] -> attnS[k][py*8+px]
    for (int t = tid; t < 144; t += 128) {
        const int k   = t >> 4;
        const int rem = t & 15;
        const int py  = rem >> 1;
        const int h4  = rem & 1;          // which float4 of the 8-wide row
        const float4 v = *(const float4*)(prob +
            ((((size_t)b * 9 + k) * 512 + gy * 8 + py) * 512 + gx * 8 + h4 * 4));
        *(float4*)&attnS[k * LDST + py * 8 + h4 * 4] = v;
    }
    // ---- stage kv_merge: replicate-clamped 3x3 neighbors of fea -> kvmS[k][c]
    for (int t = tid; t < 144; t += 128) {
        const int k  = t >> 4;
        const int c4 = (t & 15) << 2;
        const int dy = k / 3 - 1, dx = k % 3 - 1;
        int ny = gy + dy; ny = ny < 0 ? 0 : (ny > 63 ? 63 : ny);
        int nx = gx + dx; nx = nx < 0 ? 0 : (nx > 63 ? 63 : nx);
        const int n = ny * 64 + nx;
        const float4 v = *(const float4*)(fea + (((size_t)b * 4096 + n) * 64 + c4));
        *(float4*)&kvmS[k * LDST + c4] = v;
    }
    // ---- zero K-pad rows 9..11 (K chunk 2 reads K = 8..11)
    for (int i = tid; i < 3 * LDST; i += 128) {
        attnS[9 * LDST + i] = 0.0f;
        kvmS [9 * LDST + i] = 0.0f;
    }
    __syncthreads();

    const int wv   = tid >> 5;            // wave id: M-strip 0..3
    const int lane = tid & 31;
    const int l16  = lane & 15;
    const int hi   = lane >> 4;           // lane half selects K {0,1} vs {2,3}
    const int pRow = wv * 16 + l16;       // A-matrix row (M) held by this lane

    // A fragments for the 3 K-chunks (16x4 fp32 layout: V0=K0/K2, V1=K1/K3)
    v2f a[3];
#pragma unroll
    for (int kc = 0; kc < 3; ++kc) {
        const int k0 = kc * 4 + hi * 2;
        a[kc].x = attnS[ k0      * LDST + pRow];
        a[kc].y = attnS[(k0 + 1) * LDST + pRow];
    }

    v8f acc[4] = {};
#pragma unroll
    for (int nt = 0; nt < 4; ++nt) {
        const int col = nt * 16 + l16;    // B-matrix column (N) held by this lane
#pragma unroll
        for (int kc = 0; kc < 3; ++kc) {
            const int k0 = kc * 4 + hi * 2;
            v2f bb;
            bb.x = kvmS[ k0      * LDST + col];
            bb.y = kvmS[(k0 + 1) * LDST + col];
            acc[nt] = __builtin_amdgcn_wmma_f32_16x16x4_f32(
                /*neg_a=*/false, a[kc], /*neg_b=*/false, bb,
                /*c_mod=*/(short)0, acc[nt], /*reuse_a=*/false, /*reuse_b=*/false);
        }
    }

    // ---- D fragments -> outS[c][p]  (C/D layout: VGPR r = rows r / r+8)
#pragma unroll
    for (int nt = 0; nt < 4; ++nt) {
        const int c = nt * 16 + l16;
#pragma unroll
        for (int r = 0; r < 8; ++r) {
            outS[c * LDST + wv * 16 + hi * 8 + r] = acc[nt][r];
        }
    }
    __syncthreads();

    // ---- coalesced float4 stores: out[b, c, gy*8+py, gx*8+px]
    const size_t outBase = (size_t)b * 64 * 512 * 512;
    for (int t = tid; t < 1024; t += 128) {
        const int c   = t >> 4;
        const int rem = t & 15;
        const int py  = rem >> 1;
        const int h4  = rem & 1;
        const float4 v = *(const float4*)&outS[c * LDST + py * 8 + h4 * 4];
        *(float4*)(out + outBase +
                   (((size_t)c * 512) + gy * 8 + py) * 512 + gx * 8 + h4 * 4) = v;
    }
}

extern "C" void kernel_launch(void* const* d_in, const int* in_sizes, int n_in,
                              void* d_out, int out_size, void* d_ws, size_t ws_size,
                              hipStream_t stream) {
    (void)in_sizes; (void)n_in; (void)out_size; (void)d_ws; (void)ws_size;
    const float* fea  = (const float*)d_in[0];   // [4, 4096, 64]
    const float* prob = (const float*)d_in[1];   // [4, 9, 512, 512]
    float* out = (float*)d_out;                  // [4, 64, 512, 512]
    dim3 grid(4 * 64 * 64);   // one block per (b, gy, gx) grid cell
    dim3 block(128);          // 4 wave32 waves
    hipLaunchKernelGGL(sp_upsample_wmma, grid, block, 0, stream, fea, prob, out);
}